// Aggr_50637664420290
// MI455X (gfx1250) — compile-verified
//
#include <hip/hip_runtime.h>
#include <hip/hip_bf16.h>

typedef __attribute__((ext_vector_type(2))) float v2f;
typedef __attribute__((ext_vector_type(8))) float v8f;

#define THREADS 256

// ---------------------------------------------------------------- init: zero degree counters and output
__global__ void gcn_init_kernel(unsigned int* __restrict__ cnt, float* __restrict__ out,
                                int N, int outN) {
    int i = blockIdx.x * blockDim.x + threadIdx.x;
    if (i < N) cnt[i] = 0u;
    if (i < outN) out[i] = 0.0f;
}

// ---------------------------------------------------------------- in-degree count at col (u32 atomics, exact)
__global__ void gcn_count_kernel(const int* __restrict__ col, unsigned int* __restrict__ cnt,
                                 int E, int vec4ok) {
    int t = blockIdx.x * blockDim.x + threadIdx.x;
    int e = t * 4;
    if (e >= E) return;
    if (vec4ok && (e + 3) < E) {
        __builtin_prefetch(col + e + 8192, 0, 1);   // global_prefetch_b8 on the sequential stream
        int4 c4 = *(const int4*)(col + e);
        atomicAdd(&cnt[c4.x], 1u);
        atomicAdd(&cnt[c4.y], 1u);
        atomicAdd(&cnt[c4.z], 1u);
        atomicAdd(&cnt[c4.w], 1u);
    } else {
        int lim = (e + 4 < E) ? (e + 4) : E;
        for (int k = e; k < lim; ++k) atomicAdd(&cnt[col[k]], 1u);
    }
}

// ---------------------------------------------------------------- dinv = rsqrt(indeg + 1); copy h into ws
__global__ void gcn_dinv_kernel(const unsigned int* __restrict__ cnt,
                                const float* __restrict__ h_in,
                                float* __restrict__ dinv, float* __restrict__ h0, int N) {
    int i = blockIdx.x * blockDim.x + threadIdx.x;
    if (i < N) {
        dinv[i] = rsqrtf((float)cnt[i] + 1.0f);   // deg >= 1 always (self loop)
        h0[i]   = h_in[i];
    }
}

// ---------------------------------------------------------------- per layer: pack (dinv,h) and seed hn with
// the self-loop term  hn[i] = dinv[i]^2 * h[i]
__global__ void gcn_prep_kernel(const float* __restrict__ dinv, const float* __restrict__ hcur,
                                float2* __restrict__ g, float* __restrict__ hn, int N) {
    int i = blockIdx.x * blockDim.x + threadIdx.x;
    if (i < N) {
        float d = dinv[i];
        float h = hcur[i];
        g[i]  = make_float2(d, h);
        hn[i] = d * d * h;
    }
}

// ---------------------------------------------------------------- edge scatter: hn[c] += dinv[r]*dinv[c]*h[r]
__global__ void gcn_edge_kernel(const int* __restrict__ row, const int* __restrict__ col,
                                const float2* __restrict__ g, const float* __restrict__ dinv,
                                float* __restrict__ hn, int E, int vec4ok) {
    int t = blockIdx.x * blockDim.x + threadIdx.x;
    int e = t * 4;
    if (e >= E) return;
    if (vec4ok && (e + 3) < E) {
        __builtin_prefetch(row + e + 8192, 0, 1);
        __builtin_prefetch(col + e + 8192, 0, 1);
        int4 r4 = *(const int4*)(row + e);          // b128 sequential edge stream
        int4 c4 = *(const int4*)(col + e);
        float2 g0 = g[r4.x];                        // b64 gather (L2-resident, 0.8 MB array)
        float2 g1 = g[r4.y];
        float2 g2 = g[r4.z];
        float2 g3 = g[r4.w];
        float d0 = dinv[c4.x];
        float d1 = dinv[c4.y];
        float d2 = dinv[c4.z];
        float d3 = dinv[c4.w];
        unsafeAtomicAdd(&hn[c4.x], g0.x * g0.y * d0);   // native global_atomic_add_f32
        unsafeAtomicAdd(&hn[c4.y], g1.x * g1.y * d1);
        unsafeAtomicAdd(&hn[c4.z], g2.x * g2.y * d2);
        unsafeAtomicAdd(&hn[c4.w], g3.x * g3.y * d3);
    } else {
        int lim = (e + 4 < E) ? (e + 4) : E;
        for (int k = e; k < lim; ++k) {
            int r = row[k], c = col[k];
            float2 gr = g[r];
            unsafeAtomicAdd(&hn[c], gr.x * gr.y * dinv[c]);
        }
    }
}

// ---------------------------------------------------------------- sum(h^2) with a WMMA-based wave32 reduction
// a = {s, 0} per lane => A[m][0]=s[m], A[m][2]=s[m+16], K1=K3=0.  B = ones.
// D[m][n] = s[m] + s[m+16].  Per-lane sum of the 8 D VGPRs folds 16 rows;
// lanes 0-15 hold sum(m=0..7,16..23), lanes 16-31 hold sum(m=8..15,24..31);
// one shfl_xor(16) completes the 32-lane total.  fp32-exact (RNE), EXEC all-ones.
__global__ void gcn_reduce_kernel(const float* __restrict__ h, int N, float* __restrict__ out) {
    float s = 0.0f;
    int stride = gridDim.x * blockDim.x;
    for (int i = blockIdx.x * blockDim.x + threadIdx.x; i < N; i += stride) {
        float v = h[i];
        s = fmaf(v, v, s);
    }
    v2f a; a.x = s;    a.y = 0.0f;
    v2f b; b.x = 1.0f; b.y = 1.0f;
    v8f c = {};
    c = __builtin_amdgcn_wmma_f32_16x16x4_f32(
            /*neg_a=*/false, a, /*neg_b=*/false, b,
            /*c_mod=*/(short)0, c, /*reuse_a=*/false, /*reuse_b=*/false);
    float p = c[0] + c[1] + c[2] + c[3] + c[4] + c[5] + c[6] + c[7];
    p += __shfl_xor(p, 16, 32);          // combine the two half-wave partials

    __shared__ float red[8];             // blockDim.x == 256 -> 8 waves
    int lane = threadIdx.x & 31;
    int wave = threadIdx.x >> 5;
    if (lane == 0) red[wave] = p;
    __syncthreads();
    if (threadIdx.x == 0) {
        float t = 0.0f;
        #pragma unroll
        for (int w = 0; w < 8; ++w) t += red[w];
        unsafeAtomicAdd(out, t);
    }
}

// ----------------------------------------------------------------
extern "C" void kernel_launch(void* const* d_in, const int* in_sizes, int n_in,
                              void* d_out, int out_size, void* d_ws, size_t ws_size,
                              hipStream_t stream) {
    const int N = in_sizes[0];          // 100000 nodes (h is N x 1)
    const int E = in_sizes[1] / 2;      // 6.4M edges; edge_index is (2,E) row-major
    const float* h_in = (const float*)d_in[0];
    const int*   row  = (const int*)d_in[1];        // edge_index[0] = sources
    const int*   col  = row + E;                    // edge_index[1] = targets
    float* out = (float*)d_out;

    // Workspace carve: cnt[N] u32 | dinv[N] | hA[N] | hB[N] | g[N] float2  (24*N bytes ~ 2.4 MB)
    unsigned int* cnt = (unsigned int*)d_ws;
    float* dinv = (float*)(cnt + N);
    float* hA   = dinv + N;
    float* hB   = hA + N;
    float2* g   = (float2*)(hB + N);

    const int vec4ok = ((E & 3) == 0) ? 1 : 0;      // int4 path needs col = row+E 16B-aligned
    const int gN  = (N + THREADS - 1) / THREADS;
    const int gE4 = ((E + 3) / 4 + THREADS - 1) / THREADS;

    gcn_init_kernel<<<gN, THREADS, 0, stream>>>(cnt, out, N, out_size);
    gcn_count_kernel<<<gE4, THREADS, 0, stream>>>(col, cnt, E, vec4ok);
    gcn_dinv_kernel<<<gN, THREADS, 0, stream>>>(cnt, h_in, dinv, hA, N);

    float* hc = hA;
    float* hn = hB;
    const int L = out_size;             // 10 layers
    for (int l = 0; l < L; ++l) {
        gcn_prep_kernel<<<gN, THREADS, 0, stream>>>(dinv, hc, g, hn, N);
        gcn_edge_kernel<<<gE4, THREADS, 0, stream>>>(row, col, g, dinv, hn, E, vec4ok);
        gcn_reduce_kernel<<<256, THREADS, 0, stream>>>(hn, N, out + l);
        float* tmp = hc; hc = hn; hn = tmp;
    }
}